// SoftAUC_1717986918748
// MI455X (gfx1250) — compile-verified
//
#include <hip/hip_runtime.h>
#include <hip/hip_bf16.h>

typedef float v2f __attribute__((ext_vector_type(2)));
typedef float v8f __attribute__((ext_vector_type(8)));

#define BS 16384
#define JT 2048   // j-tile staged in LDS (float2 -> 16KB)

// Phase 0: half-scores = 0.5*(logits[:,1]-logits[:,0]); pack {hscore, pos}.
// Half-scaling lets the pair kernel use sigma(x) = 0.5 + 0.5*tanh(x/2)
// with a single subtract feeding v_tanh_f32.
__global__ void __launch_bounds__(256)
prep_kernel(const float* __restrict__ logits,
            const unsigned char* __restrict__ labels,
            float2* __restrict__ sp) {
  int i = blockIdx.x * blockDim.x + threadIdx.x;
  if (i < BS) {
    float hs = 0.5f * (logits[2 * i + 1] - logits[2 * i]);
    float p  = labels[i] ? 1.0f : 0.0f;
    sp[i] = make_float2(hs, p);
  }
}

// masked sigmoid contribution: 2w * sigma(2*hd) where w = 0.5*neg_i*pos_j,
// a = w * tanh(hd) + w  ==  neg_i * pos_j * sigma(s_j - s_i)
__device__ __forceinline__ float pair_val(float w, float hd) {
#if __has_builtin(__builtin_amdgcn_tanhf)
  return __builtin_fmaf(w, __builtin_amdgcn_tanhf(hd), w);   // 1 trans + 1 fma
#else
  return (2.0f * w) * __builtin_amdgcn_rcpf(1.0f + __expf(-2.0f * hd));
#endif
}

// Phase 1: each wave owns 16 i-rows, sweeps all j through an LDS tile.
// Masked sigmoid values form a 16x4 f32 A-matrix; WMMA with all-ones B
// accumulates row sums into a 16x16 f32 C entirely in the matrix pipe,
// co-executing with the VALU/trans sigmoid stream.
__global__ void __launch_bounds__(256)
pairs_kernel(const float2* __restrict__ sp, float* __restrict__ partials) {
  __shared__ __align__(16) float2 tile[JT];

  const int tid    = threadIdx.x;
  const int lane   = tid & 31;
  const int waveId = blockIdx.x * 8 + (tid >> 5);   // 0..1023
  const int i      = waveId * 16 + (lane & 15);     // this lane's i-row
  const int koff   = (lane >> 4) << 1;              // lanes 0-15: K=0,1; 16-31: K=2,3

  const float2 mi    = sp[i];
  const float  hsi   = mi.x;                        // 0.5 * s_i
  const float  neghf = 0.5f - 0.5f * mi.y;          // 0.5 * neg_i

  v8f c = {};                                       // 16x16 f32 accumulator
  v2f b = {1.0f, 1.0f};                             // all-ones 4x16 B (layout-agnostic)

  for (int chunk = 0; chunk < BS; chunk += JT) {
    __syncthreads();
    for (int k = tid; k < JT; k += 256) tile[k] = sp[chunk + k];
    __syncthreads();

    for (int jj = 0; jj < JT; jj += 4) {
      // two adjacent {0.5*s_j, pos_j} pairs -> one aligned 16B LDS read
      const float4 q = *reinterpret_cast<const float4*>(&tile[jj + koff]);
      float w0 = neghf * q.y;                       // 0.5*neg_i*pos_j
      float w1 = neghf * q.w;
      float a0 = pair_val(w0, q.x - hsi);           // hd = 0.5*(s_j - s_i)
      float a1 = pair_val(w1, q.z - hsi);
      v2f a = {a0, a1};
      // D = A x Ones + C : accumulates Sum_k A[m][k] into every column of C
      c = __builtin_amdgcn_wmma_f32_16x16x4_f32(false, a, false, b,
                                                (short)0, c, false, false);
    }
  }

  // All columns of C are identical row-sums. Column N=0 lives in lane 0
  // (M=0..7 across the 8 VGPRs) and lane 16 (M=8..15).
  float t = c[0] + c[1] + c[2] + c[3] + c[4] + c[5] + c[6] + c[7];
  float total = __shfl(t, 0, 32) + __shfl(t, 16, 32);
  if (lane == 0) partials[waveId] = total;
}

// Phase 2: deterministic fixed-order reduction + final scalar.
__global__ void __launch_bounds__(256)
finalize_kernel(const float2* __restrict__ sp, const float* __restrict__ partials,
                float* __restrict__ out) {
  __shared__ float rednp[256];
  __shared__ float redps[256];
  const int tid = threadIdx.x;

  float np = 0.0f;
  for (int k = tid; k < BS; k += 256) np += sp[k].y;
  float ps = 0.0f;
  for (int p = tid; p < 1024; p += 256) ps += partials[p];

  rednp[tid] = np;
  redps[tid] = ps;
  __syncthreads();
  for (int s = 128; s > 0; s >>= 1) {
    if (tid < s) { rednp[tid] += rednp[tid + s]; redps[tid] += redps[tid + s]; }
    __syncthreads();
  }
  if (tid == 0) {
    float numpos = rednp[0];
    float numneg = (float)BS - numpos;
    float denom  = numpos * numneg;
    out[0] = (denom > 0.0f) ? (1.0f - redps[0] / denom) : 0.0f;
  }
}

extern "C" void kernel_launch(void* const* d_in, const int* in_sizes, int n_in,
                              void* d_out, int out_size, void* d_ws, size_t ws_size,
                              hipStream_t stream) {
  const float* logits         = (const float*)d_in[0];
  const unsigned char* labels = (const unsigned char*)d_in[1];

  float2* sp       = (float2*)d_ws;                                  // 16384 float2
  float*  partials = (float*)((char*)d_ws + BS * sizeof(float2));    // 1024 floats

  prep_kernel<<<BS / 256, 256, 0, stream>>>(logits, labels, sp);
  pairs_kernel<<<(BS / 16) / 8, 256, 0, stream>>>(sp, partials);     // 128 blocks x 8 waves
  finalize_kernel<<<1, 256, 0, stream>>>(sp, partials, (float*)d_out);
}